// DetectionLoss_60473139528033
// MI455X (gfx1250) — compile-verified
//
#include <hip/hip_runtime.h>
#include <hip/hip_bf16.h>
#include <math.h>

#define B_  32
#define NQ_ 1024
#define C_  1024
#define M_  128
#define D_  2

typedef float v2f __attribute__((ext_vector_type(2)));
typedef float v8f __attribute__((ext_vector_type(8)));

__device__ __forceinline__ float softplus_f(float x) {
    // stable: max(x,0) + log1p(exp(-|x|))
    return fmaxf(x, 0.0f) + log1pf(expf(-fabsf(x)));
}

// ---------------------------------------------------------------------------
// Kernel 1: one wave32 per matched (b, i) pair.  4096 rows total.
//   - log-softmax CE over the gathered logit row (1024 fp32, coalesced float4)
//   - L1 centroid term, softplus(-conf) obj term, softplus(conf) matched term
// ---------------------------------------------------------------------------
__global__ __launch_bounds__(256) void detloss_rows_kernel(
    const float* __restrict__ pred_centroids,  // (B, NQ, D)
    const float* __restrict__ pred_logits,     // (B, NQ, C)
    const float* __restrict__ pred_conf,       // (B, NQ)
    const float* __restrict__ gt_centroids,    // (B, M, D)
    const int*   __restrict__ gt_classes,      // (B, M)
    const int*   __restrict__ pred_idx,        // (B, M)
    const int*   __restrict__ gt_idx,          // (B, M)
    float* __restrict__ ws_pos,                // (B*M)
    float* __restrict__ ws_ce,                 // (B*M)
    float* __restrict__ ws_obj,                // (B*M)
    float* __restrict__ ws_msp)                // (B*M)
{
    const int lane = threadIdx.x & 31;
    const int wave = threadIdx.x >> 5;
    const int r = blockIdx.x * 8 + wave;       // 0 .. 4095
    const int b = r >> 7;                      // r / M_
    const int i = r & (M_ - 1);

    const int pidx = pred_idx[b * M_ + i];
    const int gidx = gt_idx[b * M_ + i];
    const int y    = gt_classes[b * M_ + gidx];

    const float* base = pred_logits + (size_t)(b * NQ_ + pidx) * C_;
    const float4* b4  = reinterpret_cast<const float4*>(base);

    // each lane holds 32 logits (8 x float4), coalesced 32-lane-wide stream (128 floats per step)
    float4 x[8];
#pragma unroll
    for (int j = 0; j < 8; ++j) x[j] = b4[j * 32 + lane];

    float mx = -3.402823466e38f;
#pragma unroll
    for (int j = 0; j < 8; ++j)
        mx = fmaxf(mx, fmaxf(fmaxf(x[j].x, x[j].y), fmaxf(x[j].z, x[j].w)));
#pragma unroll
    for (int off = 16; off >= 1; off >>= 1)
        mx = fmaxf(mx, __shfl_xor(mx, off, 32));

    float s = 0.0f;
#pragma unroll
    for (int j = 0; j < 8; ++j) {
        s += expf(x[j].x - mx);
        s += expf(x[j].y - mx);
        s += expf(x[j].z - mx);
        s += expf(x[j].w - mx);
    }
#pragma unroll
    for (int off = 16; off >= 1; off >>= 1)
        s += __shfl_xor(s, off, 32);

    if (lane == 0) {
        const float lse = mx + logf(s);
        const float ly  = base[y];             // target-class logit (L0/L2 hit)
        ws_ce[r] = lse - ly;

        const float px0 = pred_centroids[(size_t)(b * NQ_ + pidx) * D_ + 0];
        const float px1 = pred_centroids[(size_t)(b * NQ_ + pidx) * D_ + 1];
        const float gx0 = gt_centroids[(b * M_ + gidx) * D_ + 0];
        const float gx1 = gt_centroids[(b * M_ + gidx) * D_ + 1];
        ws_pos[r] = fabsf(px0 - gx0) + fabsf(px1 - gx1);

        const float cm = pred_conf[b * NQ_ + pidx];
        ws_obj[r] = softplus_f(-cm);
        ws_msp[r] = softplus_f(cm);
    }
}

// ---------------------------------------------------------------------------
// Kernel 2: single 1024-thread block, fully deterministic reduction.
//   - pos / obj / matched-softplus / all-softplus sums via LDS tree (float4)
//   - CE sum via V_WMMA_F32_16X16X4_F32 with an all-ones B matrix:
//       D = A x 1 + C  => fp32 row sums.  Grand total of D over all 256
//       slots equals 16 * (sum of A elements), independent of VGPR layout.
// ---------------------------------------------------------------------------
__global__ __launch_bounds__(1024) void detloss_reduce_kernel(
    const float* __restrict__ pred_conf,   // (B*NQ) = 32768
    const float* __restrict__ ws_pos,
    const float* __restrict__ ws_ce,
    const float* __restrict__ ws_obj,
    const float* __restrict__ ws_msp,
    float* __restrict__ out)               // 6 floats
{
    __shared__ float4 red[1024];
    __shared__ float  wsum[32];

    const int t = threadIdx.x;

    float a_pos = 0.f, a_obj = 0.f, a_msp = 0.f, a_conf = 0.f;
#pragma unroll
    for (int k = 0; k < 4; ++k) {
        const int idx = t + 1024 * k;
        a_pos += ws_pos[idx];
        a_obj += ws_obj[idx];
        a_msp += ws_msp[idx];
    }
#pragma unroll
    for (int k = 0; k < 32; ++k)
        a_conf += softplus_f(pred_conf[t + 1024 * k]);

    red[t] = make_float4(a_pos, a_obj, a_msp, a_conf);
    __syncthreads();
#pragma unroll
    for (int sft = 512; sft >= 1; sft >>= 1) {
        if (t < sft) {
            float4 m = red[t];
            const float4 o = red[t + sft];
            m.x += o.x; m.y += o.y; m.z += o.z; m.w += o.w;
            red[t] = m;
        }
        __syncthreads();
    }

    // ---- CE reduction: wave w reduces ws_ce[w*128 .. w*128+127] ----
    const int lane = t & 31;
    const int w    = t >> 5;

    float ce_wave;
#if __has_builtin(__builtin_amdgcn_wmma_f32_16x16x4_f32)
    {
        v8f acc = {};
        const v2f ones = {1.0f, 1.0f};
#pragma unroll
        for (int ch = 0; ch < 2; ++ch) {
            const float* p = ws_ce + w * 128 + ch * 64 + lane * 2;
            v2f a = {p[0], p[1]};
            acc = __builtin_amdgcn_wmma_f32_16x16x4_f32(
                /*neg_a=*/false, a, /*neg_b=*/false, ones,
                /*c_mod=*/(short)0, acc, /*reuse_a=*/false, /*reuse_b=*/false);
        }
        float sl = 0.f;
#pragma unroll
        for (int q = 0; q < 8; ++q) sl += acc[q];
#pragma unroll
        for (int off = 16; off >= 1; off >>= 1)
            sl += __shfl_xor(sl, off, 32);
        ce_wave = sl * 0.0625f;                 // / 16 columns (exact pow2)
    }
#else
    {
        const float* p = ws_ce + w * 128;
        float sl = p[lane * 4 + 0] + p[lane * 4 + 1] + p[lane * 4 + 2] + p[lane * 4 + 3];
#pragma unroll
        for (int off = 16; off >= 1; off >>= 1)
            sl += __shfl_xor(sl, off, 32);
        ce_wave = sl;
    }
#endif
    if (lane == 0) wsum[w] = ce_wave;
    __syncthreads();

    if (t == 0) {
        float S_ce = 0.f;
#pragma unroll
        for (int k = 0; k < 32; ++k) S_ce += wsum[k];
        const float4 S = red[0];
        const float S_pos = S.x, S_obj = S.y, S_msp = S.z, S_all = S.w;

        const float invB = 1.0f / (float)B_;
        const float lp = 5.0f * (S_pos / (float)(M_ * D_)) * invB;
        const float lc = 1.0f * (S_ce  / (float)M_)        * invB;
        const float lo = 2.0f * (S_obj / (float)M_)        * invB;
        const float ln = 0.1f * ((S_all - S_msp) / (float)(NQ_ - M_)) * invB;

        out[0] = lp;
        out[1] = lc;
        out[2] = lo;
        out[3] = ln;
        out[4] = lp + lc + lo + ln;
        out[5] = (float)M_;
    }
}

extern "C" void kernel_launch(void* const* d_in, const int* in_sizes, int n_in,
                              void* d_out, int out_size, void* d_ws, size_t ws_size,
                              hipStream_t stream) {
    const float* pred_centroids = (const float*)d_in[0];
    const float* pred_logits    = (const float*)d_in[1];
    const float* pred_conf      = (const float*)d_in[2];
    const float* gt_centroids   = (const float*)d_in[3];
    const int*   gt_classes     = (const int*)d_in[4];
    const int*   pred_idx       = (const int*)d_in[5];
    const int*   gt_idx         = (const int*)d_in[6];
    float* out = (float*)d_out;

    float* ws     = (float*)d_ws;
    float* ws_pos = ws;
    float* ws_ce  = ws + 4096;
    float* ws_obj = ws + 8192;
    float* ws_msp = ws + 12288;

    // 4096 rows, 8 waves (one row each) per 256-thread block
    detloss_rows_kernel<<<512, 256, 0, stream>>>(
        pred_centroids, pred_logits, pred_conf, gt_centroids,
        gt_classes, pred_idx, gt_idx,
        ws_pos, ws_ce, ws_obj, ws_msp);

    detloss_reduce_kernel<<<1, 1024, 0, stream>>>(
        pred_conf, ws_pos, ws_ce, ws_obj, ws_msp, out);
}